// E2SRC_Module_3092376453877
// MI455X (gfx1250) — compile-verified
//
#include <hip/hip_runtime.h>

// ---------------------------------------------------------------------------
// Event-histogram (segment_sum) kernel for MI455X / gfx1250.
//
// N = 16,777,216 events x float4 (t, x, y, p) -> two f32 histograms of
// 614,400 bins, interleaved into a 1,228,800-float output.
//
// Bound by streaming 268 MB of input once (~11.5 us @ 23.3 TB/s HBM) plus
// 33.5M L2-resident f32 atomics (4.9 MB histogram << 192 MB L2).
// CDNA5 paths used: non-temporal global_load_b128 (TH_LOAD_NT) for the
// single-use input stream, and native non-returning global_atomic_add_f32
// at device scope via inline asm (STOREcnt path, no CAS fallback).
// ---------------------------------------------------------------------------

typedef float f32x4 __attribute__((ext_vector_type(4)));

#define PW_N       20          // (W+... ) PW = 20 patches wide
#define TOKEN_NUM  300         // PH*PW = 15*20
#define PATCH_SIZE 256         // 16*16
#define TIME_DIV   4
#define SLICE      (PATCH_SIZE * TOKEN_NUM)   // 76800
#define TOTAL      (TIME_DIV * 2 * SLICE)     // 614400

// Zero-fill d_out (harness poisons it with 0xAA before timing).
__global__ void __launch_bounds__(256) zero_kernel(float* __restrict__ out, int n4) {
    f32x4 z = {0.0f, 0.0f, 0.0f, 0.0f};
    int stride = gridDim.x * blockDim.x;
    for (int i = blockIdx.x * blockDim.x + threadIdx.x; i < n4; i += stride)
        ((f32x4*)out)[i] = z;
}

// Non-returning device-scope f32 atomic add -> global_atomic_add_f32.
// Inline asm guarantees the native CDNA5 instruction (no CAS-loop fallback)
// and the non-returning form, which only touches STOREcnt (never waited on;
// S_ENDPGM does the implicit wait-idle).
__device__ __forceinline__ void gatomic_add_f32(float* p, float v) {
    asm volatile("global_atomic_add_f32 %0, %1, off scope:SCOPE_DEV"
                 :: "v"(p), "v"(v)
                 : "memory");
}

__global__ void __launch_bounds__(256) e2src_hist_kernel(const f32x4* __restrict__ x,
                                                         float* __restrict__ out,
                                                         int n) {
    // Compile-time bin constants, computed exactly as the Python reference
    // (double precision host math, then used as f32).
    const float cW = (float)(319.0 / 20.0 + 1e-4);   // W/PW + B = 15.9501
    const float cH = (float)(239.0 / 15.0 + 1e-4);   // H/PH + B

    // Uniform endpoints -> scalar s_load (uniform address), L2-broadcast.
    const float t0 = x[0].x;
    const float tl = x[n - 1].x;
    const float denW = tl - t0 + 1e-4f;   // for wt
    const float denT = tl - t0 + 1.0f;    // for dtime

    int stride = gridDim.x * blockDim.x;
    for (int i = blockIdx.x * blockDim.x + threadIdx.x; i < n; i += stride) {
        // One event per lane: coalesced 128-bit streaming load, NT hint
        // (input is touched exactly once -> don't pollute WGP$/L2).
        f32x4 e = __builtin_nontemporal_load(&x[i]);
        float t = e.x, xs = e.y, ys = e.z, p = e.w;

        float w  = (p != 2.0f) ? 1.0f : 0.0f;
        float wt = (t - t0) / denW;

        // jnp.mod(a, c) == a - floor(a/c)*c for positive operands; reuse the
        // floor-div for both the patch coordinate and the in-patch token.
        float qx = floorf(xs / cW);
        float mx = floorf(xs - qx * cW);
        float qy = floorf(ys / cH);
        float my = floorf(ys - qy * cH);

        float posf = qx + qy * (float)PW_N;        // patch index
        float tokf = mx + my * 16.0f;              // in-patch token ((W+1)//PW == 16)
        float df   = floorf((4.0f * (t - t0)) / denT);

        // Reference clips to [0, bins] INCLUSIVE (jnp.clip upper == bins).
        int di = min(max((int)df,   0), TIME_DIV);
        int pi = min(max((int)p,    0), 2);
        int ti = min(max((int)tokf, 0), PATCH_SIZE);
        int si = min(max((int)posf, 0), TOKEN_NUM);

        int l = (di * 2 + pi) * SLICE + ti * TOKEN_NUM + si;

        // JAX segment_sum drops out-of-range indices (possible after the
        // inclusive clip): replicate that.
        if ((unsigned)l < (unsigned)TOTAL) {
            int q = l / SLICE;              // (dtime*2 + p) slice id, [0,8)
            int r = l - q * SLICE;          // tok*300 + pos
            // Output layout: y[d][p][c][tok][pos], c=0 -> w, c=1 -> wt.
            float* base = out + q * (2 * SLICE) + r;
            gatomic_add_f32(base,         w);   // c = 0
            gatomic_add_f32(base + SLICE, wt);  // c = 1
        }
    }
}

extern "C" void kernel_launch(void* const* d_in, const int* in_sizes, int n_in,
                              void* d_out, int out_size, void* d_ws, size_t ws_size,
                              hipStream_t stream) {
    (void)n_in; (void)d_ws; (void)ws_size;

    const f32x4* x = (const f32x4*)d_in[0];
    float* out = (float*)d_out;

    int n  = in_sizes[0] / 4;   // number of events (16,777,216)
    int n4 = out_size / 4;      // output float4 count (307,200)

    // 1) zero the histogram output
    int zblocks = (n4 + 255) / 256;
    if (zblocks > 1200) zblocks = 1200;
    zero_kernel<<<zblocks, 256, 0, stream>>>(out, n4);

    // 2) scatter-accumulate: 2M threads, grid-stride (8 events/thread),
    //    consecutive lanes hit consecutive 16B records -> b128 coalescing.
    e2src_hist_kernel<<<8192, 256, 0, stream>>>(x, out, n);
}